// SelfConnectionIntro_35227321762138
// MI455X (gfx1250) — compile-verified
//
#include <hip/hip_runtime.h>
#include <math.h>

// ---------------------------------------------------------------------------
// Grouped-by-species fp32 WMMA GEMM for the SelfConnection einsum.
//   out[z] = concat( x0[z]·W0[:,s,:]*a0, (x1[z,:,i]·W1[:,s,:])*a1, (x2..)*a2 )
// Pipeline: init -> classify(one-hot argmax)+histogram -> scan/build tiles ->
//           scatter (group nodes by species) -> grouped GEMM:
//             * TDM tensor_load_to_lds stages W0/W1/W2 species slices
//               (2D strided tile, pad_enable for bank-conflict-free strides)
//             * V_WMMA_F32_16X16X4_F32 on 16-node tiles, A frags as ds_load_b64
//             * outputs staged in LDS, coalesced float4 stores
// Workspace use: ~168 KB of d_ws (ints).
// ---------------------------------------------------------------------------

#define N_NODES   20000
#define N_SPECIES 89
#define DIM_X     480
#define MAX_TILES 1344     // >= sum_s ceil(count_s/16) <= 1250 + 89 = 1339

// workspace layout (int offsets)
#define WS_NUMTILES 0
#define WS_COUNT    64
#define WS_CURSOR   192
#define WS_OFFSETS  320              // 90 entries
#define WS_SPEC     448
#define WS_SORTED   (WS_SPEC + N_NODES)
#define WS_TILES    (WS_SORTED + N_NODES)

typedef __attribute__((ext_vector_type(2))) float        v2f;
typedef __attribute__((ext_vector_type(8))) float        v8f;
typedef __attribute__((ext_vector_type(4))) unsigned int u32x4;
typedef __attribute__((ext_vector_type(4))) int          i32x4;
typedef __attribute__((ext_vector_type(8))) int          i32x8;

// LDS strides in dwords (bank-conflict-free fragment access, see analysis)
#define XS0  132   // x0 plane:  16 x 128, stride%64==4
#define XS1  68    // x1 planes: 3 x 16 x 64, stride%64==4
#define XS2  36    // x2 planes: 5 x 16 x 32, stride%64==36
#define OTS  484   // out tile:  16 x 480, stride%64==36, 16B-aligned rows
#define W0S  144   // 128 + 16 pad dwords (TDM pad)  -> %64 == 16
#define W1S  80    // 64 + 16                        -> %64 == 16
#define W2S  48    // 32 + 16                        -> %64 == 48

// ---------------------------------------------------------------------------

__global__ void sci_init(int* __restrict__ ws) {
  int t = threadIdx.x;
  if (t < N_SPECIES) { ws[WS_COUNT + t] = 0; ws[WS_CURSOR + t] = 0; }
  if (t == 0) ws[WS_NUMTILES] = 0;
}

__global__ __launch_bounds__(256) void sci_classify(const float* __restrict__ attr,
                                                    int* __restrict__ ws) {
  int z = blockIdx.x * 256 + threadIdx.x;
  if (z >= N_NODES) return;
  const float* row = attr + (size_t)z * N_SPECIES;
  int s = 0;
  for (int v = 0; v < N_SPECIES; ++v) {
    if (row[v] > 0.5f) { s = v; break; }
  }
  ws[WS_SPEC + z] = s;
  atomicAdd(&ws[WS_COUNT + s], 1);
}

__global__ __launch_bounds__(128) void sci_build(int* __restrict__ ws) {
  __shared__ int tbase[N_SPECIES + 1];
  if (threadIdx.x == 0) {
    int o = 0, tb = 0;
    for (int sp = 0; sp < N_SPECIES; ++sp) {
      ws[WS_OFFSETS + sp] = o;
      tbase[sp] = tb;
      int c = ws[WS_COUNT + sp];
      o  += c;
      tb += (c + 15) >> 4;
    }
    ws[WS_OFFSETS + N_SPECIES] = o;
    tbase[N_SPECIES] = tb;
    ws[WS_NUMTILES] = tb;
  }
  __syncthreads();
  int sp = threadIdx.x;
  if (sp < N_SPECIES) {
    int c    = ws[WS_COUNT + sp];
    int off  = ws[WS_OFFSETS + sp];
    int nt   = (c + 15) >> 4;
    int base = tbase[sp];
    for (int tt = 0; tt < nt; ++tt) {
      int st = off + 16 * tt;
      int cc = c - 16 * tt; if (cc > 16) cc = 16;
      // pack: start[14:0] | cnt[19:15] | species[26:20]
      ws[WS_TILES + base + tt] = st | (cc << 15) | (sp << 20);
    }
  }
}

__global__ __launch_bounds__(256) void sci_scatter(int* __restrict__ ws) {
  int z = blockIdx.x * 256 + threadIdx.x;
  if (z >= N_NODES) return;
  int s = ws[WS_SPEC + z];
  int p = ws[WS_OFFSETS + s] + atomicAdd(&ws[WS_CURSOR + s], 1);
  ws[WS_SORTED + p] = z;
}

// ---------------------------------------------------------------------------
// TDM: DMA a 2D tile (tile_w x tile_h elements of 4B) whose rows are
// row_stride elements apart in global memory, into LDS at lds_addr, inserting
// pad_amount(+1) dwords of LDS padding every 2^(pad_interval+1) dwords.
// ---------------------------------------------------------------------------
__device__ __forceinline__ void tdm_load_2d(const float* gptr, unsigned lds_addr,
                                            unsigned tile_w, unsigned tile_h,
                                            unsigned long long row_stride,
                                            unsigned pad_int_code, unsigned pad_amt_code) {
  unsigned long long ga = (unsigned long long)(uintptr_t)gptr;
  u32x4 g0;
  g0.x = 1u;                                              // count=1, user mode
  g0.y = lds_addr;                                        // LDS byte address
  g0.z = (unsigned)(ga & 0xFFFFFFFFu);                    // global addr lo
  g0.w = (unsigned)((ga >> 32) & 0x1FFFFFFu) | (2u << 30);// addr hi | type=2
  i32x8 g1;
  g1[0] = (int)((2u << 16) |                // data_size = 4B
                (1u << 20) |                // pad_enable
                (pad_int_code << 22) | (pad_amt_code << 25));
  g1[1] = (int)((tile_w & 0xFFFFu) << 16);  // abar=0 | tensor_dim0 lo
  g1[2] = (int)((tile_w >> 16) | ((tile_h & 0xFFFFu) << 16)); // td0 hi|td1 lo
  g1[3] = (int)((tile_h >> 16) | (tile_w << 16));             // td1 hi|tile_dim0
  g1[4] = (int)tile_h;                                        // tile_dim1, dim2=0
  g1[5] = (int)(unsigned)(row_stride & 0xFFFFFFFFull);        // dim0_stride lo
  g1[6] = (int)(unsigned)((row_stride >> 32) & 0xFFFFull);    // dim0_stride hi
  g1[7] = 0;
  i32x4 z4 = {0, 0, 0, 0};
#if defined(__clang_major__) && (__clang_major__ >= 23)
  i32x8 z8 = {0, 0, 0, 0, 0, 0, 0, 0};
  __builtin_amdgcn_tensor_load_to_lds(g0, g1, z4, z4, z8, 0);
#else
  __builtin_amdgcn_tensor_load_to_lds(g0, g1, z4, z4, 0);
#endif
}

__global__ __launch_bounds__(256) void sci_gemm(const float* __restrict__ x,
                                                const float* __restrict__ W0,
                                                const float* __restrict__ W1,
                                                const float* __restrict__ W2,
                                                float* __restrict__ out,
                                                const int* __restrict__ ws,
                                                float sc0, float sc1, float sc2) {
  // pool holds planar x tiles during compute, then is reused as the out tile
  __shared__ float pool[16 * XS0 + 3 * 16 * XS1 + 5 * 16 * XS2];  // 8256 dw
  __shared__ float w0s[128 * W0S];
  __shared__ float w1s[64 * W1S];
  __shared__ float w2s[32 * W2S];
  __shared__ int   nodes[16];

  const int nT   = ws[WS_NUMTILES];
  const int tile = blockIdx.x;
  if (tile >= nT) return;

  const int packed = ws[WS_TILES + tile];
  const int start = packed & 0x7FFF;
  const int cnt   = (packed >> 15) & 31;
  const int sp    = (packed >> 20) & 127;

  const int t = threadIdx.x;
  if (t < 16) {
    // tail rows duplicate row 0 (same species); their outputs are not stored
    nodes[t] = ws[WS_SORTED + start + ((t < cnt) ? t : 0)];
  }
  __syncthreads();

  // wave 0 kicks off the three weight-slice DMAs (TENSORcnt-tracked),
  // overlapping with the cooperative x-tile staging below.
  if (t < 32) {
    tdm_load_2d(W0 + (size_t)sp * 128, (unsigned)(uintptr_t)(void*)w0s,
                128, 128, 89ull * 128ull, /*2^7=128 dw*/ 6, /*16 dw*/ 15);
    tdm_load_2d(W1 + (size_t)sp * 64,  (unsigned)(uintptr_t)(void*)w1s,
                64, 64, 89ull * 64ull,   /*64 dw*/ 5, 15);
    tdm_load_2d(W2 + (size_t)sp * 32,  (unsigned)(uintptr_t)(void*)w2s,
                32, 32, 89ull * 32ull,   /*32 dw*/ 4, 15);
  }

  float* xs0 = pool;                          // 16 x XS0
  float* xs1 = pool + 16 * XS0;               // 3 planes x 16 x XS1
  float* xs2 = xs1 + 3 * 16 * XS1;            // 5 planes x 16 x XS2

  // x0 part: contiguous, float4
  for (int i = t; i < 16 * 32; i += 256) {
    int r = i >> 5, c4 = i & 31;
    float4 v = ((const float4*)(x + (size_t)nodes[r] * DIM_X))[c4];
    *(float4*)&xs0[r * XS0 + 4 * c4] = v;
  }
  // x1: de-interleave stride-3 -> planar [i][row][u]
  for (int i = t; i < 16 * 192; i += 256) {
    int r = i / 192, q = i - 192 * r;
    int u = q / 3, ii = q - 3 * u;
    xs1[(ii * 16 + r) * XS1 + u] = x[(size_t)nodes[r] * DIM_X + 128 + q];
  }
  // x2: de-interleave stride-5 -> planar [i][row][u]
  for (int i = t; i < 16 * 160; i += 256) {
    int r = i / 160, q = i - 160 * r;
    int u = q / 5, ii = q - 5 * u;
    xs2[(ii * 16 + r) * XS2 + u] = x[(size_t)nodes[r] * DIM_X + 320 + q];
  }

  if (t < 32) __builtin_amdgcn_s_wait_tensorcnt(0);
  __syncthreads();

  const int lane = t & 31;
  const int wave = t >> 5;
  const int m  = lane & 15;             // A row / B-D column
  const int kb = (lane >> 4) << 1;      // K sub-offset 0 or 2
  const int n  = m;
  const int rowBase = (lane >> 4) << 3; // D row base 0 or 8

  // 30 independent 16x16 output tiles: [0,8)=o0, [8,20)=o1, [20,30)=o2.
  // Wave w owns tiles {w, w+8, w+16, w+24}; accumulators stay in registers.
  v8f acc[4];
#pragma unroll
  for (int rep = 0; rep < 4; ++rep) {
    const int ot = wave + 8 * rep;              // wave-uniform
    if (ot < 30) {
      v8f a8 = {0.f, 0.f, 0.f, 0.f, 0.f, 0.f, 0.f, 0.f};
      if (ot < 8) {                    // o0: (16x128)·(128x16)
        const float* Ap = xs0 + m * XS0;
        const float* Bp = w0s + 16 * ot + n;
        v8f b8 = {0.f, 0.f, 0.f, 0.f, 0.f, 0.f, 0.f, 0.f};
#pragma unroll 4
        for (int k0 = 0; k0 < 64; k0 += 4) {   // two 16-deep chains
          {
            const int ka = k0 + kb;
            v2f a = *(const v2f*)(Ap + ka);
            v2f b; b.x = Bp[ka * W0S]; b.y = Bp[ka * W0S + W0S];
            a8 = __builtin_amdgcn_wmma_f32_16x16x4_f32(
                     false, a, false, b, (short)0, a8, false, false);
          }
          {
            const int ka = 64 + k0 + kb;
            v2f a = *(const v2f*)(Ap + ka);
            v2f b; b.x = Bp[ka * W0S]; b.y = Bp[ka * W0S + W0S];
            b8 = __builtin_amdgcn_wmma_f32_16x16x4_f32(
                     false, a, false, b, (short)0, b8, false, false);
          }
        }
#pragma unroll
        for (int r = 0; r < 8; ++r) a8[r] += b8[r];
      } else if (ot < 20) {            // o1: (16x64)·(64x16), planar A
        const int idx = ot - 8;
        const int ii = idx >> 2, c = idx & 3;
        const float* Ap = xs1 + (ii * 16 + m) * XS1;
        const float* Bp = w1s + 16 * c + n;
#pragma unroll 4
        for (int k0 = 0; k0 < 64; k0 += 4) {
          const int ka = k0 + kb;
          v2f a = *(const v2f*)(Ap + ka);
          v2f b; b.x = Bp[ka * W1S]; b.y = Bp[ka * W1S + W1S];
          a8 = __builtin_amdgcn_wmma_f32_16x16x4_f32(
                   false, a, false, b, (short)0, a8, false, false);
        }
      } else {                         // o2: (16x32)·(32x16), planar A
        const int idx = ot - 20;
        const int ii = idx >> 1, c = idx & 1;
        const float* Ap = xs2 + (ii * 16 + m) * XS2;
        const float* Bp = w2s + 16 * c + n;
#pragma unroll
        for (int k0 = 0; k0 < 32; k0 += 4) {
          const int ka = k0 + kb;
          v2f a = *(const v2f*)(Ap + ka);
          v2f b; b.x = Bp[ka * W2S]; b.y = Bp[ka * W2S + W2S];
          a8 = __builtin_amdgcn_wmma_f32_16x16x4_f32(
                   false, a, false, b, (short)0, a8, false, false);
        }
      }
      acc[rep] = a8;
    }
  }

  __syncthreads();   // all x reads retired -> reuse pool as the out tile

  float* ot_lds = pool;                 // 16 x OTS
#pragma unroll
  for (int rep = 0; rep < 4; ++rep) {
    const int ot = wave + 8 * rep;
    if (ot < 30) {
      v8f a8 = acc[rep];
      if (ot < 8) {
        const int c = ot;
#pragma unroll
        for (int r = 0; r < 8; ++r)
          ot_lds[(r + rowBase) * OTS + 16 * c + n] = a8[r] * sc0;
      } else if (ot < 20) {
        const int idx = ot - 8, ii = idx >> 2, c = idx & 3;
#pragma unroll
        for (int r = 0; r < 8; ++r)
          ot_lds[(r + rowBase) * OTS + 128 + 3 * (16 * c + n) + ii] = a8[r] * sc1;
      } else {
        const int idx = ot - 20, ii = idx >> 1, c = idx & 1;
#pragma unroll
        for (int r = 0; r < 8; ++r)
          ot_lds[(r + rowBase) * OTS + 320 + 5 * (16 * c + n) + ii] = a8[r] * sc2;
      }
    }
  }

  __syncthreads();

  // coalesced float4 row stores of the finished out tile
  for (int i = t; i < 16 * 120; i += 256) {
    int r = i / 120, c4 = i - 120 * r;
    if (r < cnt) {
      float4 v = *(const float4*)&ot_lds[r * OTS + 4 * c4];
      ((float4*)(out + (size_t)nodes[r] * DIM_X))[c4] = v;
    }
  }
}

extern "C" void kernel_launch(void* const* d_in, const int* in_sizes, int n_in,
                              void* d_out, int out_size, void* d_ws, size_t ws_size,
                              hipStream_t stream) {
  const float* x    = (const float*)d_in[0];
  const float* attr = (const float*)d_in[1];
  const float* W0   = (const float*)d_in[2];
  const float* W1   = (const float*)d_in[3];
  const float* W2   = (const float*)d_in[4];
  float* out = (float*)d_out;
  int*   ws  = (int*)d_ws;

  const float sc0 = (float)(1.0 / sqrt(128.0 * 89.0));
  const float sc1 = (float)(1.0 / sqrt(64.0 * 89.0));
  const float sc2 = (float)(1.0 / sqrt(32.0 * 89.0));

  const int nb = (N_NODES + 255) / 256;
  sci_init<<<1, 256, 0, stream>>>(ws);
  sci_classify<<<nb, 256, 0, stream>>>(attr, ws);
  sci_build<<<1, 128, 0, stream>>>(ws);
  sci_scatter<<<nb, 256, 0, stream>>>(ws);
  sci_gemm<<<MAX_TILES, 256, 0, stream>>>(x, W0, W1, W2, out, ws, sc0, sc1, sc2);
}